// FlexMoE_38646115729759
// MI455X (gfx1250) — compile-verified
//
#include <hip/hip_runtime.h>

#define M_DIM 4
#define B_DIM 2048
#define D_DIM 1024
#define O_DIM 1024
#define E_DIM 8
#define C_DIM 2
#define NTOK (M_DIM * B_DIM)   // 8192 tokens

typedef __attribute__((ext_vector_type(2))) float v2f;
typedef __attribute__((ext_vector_type(8))) float v8f;

// ---------------------------------------------------------------------------
// Kernel 1: gate GEMV (exact f32), winner selection, per-expert token lists.
// One wave (32 lanes) per token; coalesced reads of x; butterfly reduction.
// ---------------------------------------------------------------------------
__global__ __launch_bounds__(256) void gate_kernel(
    const float* __restrict__ x, const float* __restrict__ gate_w,
    const float* __restrict__ gate_b, int* __restrict__ counts,
    int* __restrict__ perm)
{
    const int lane = threadIdx.x & 31;
    const int wave = threadIdx.x >> 5;
    const int tok  = blockIdx.x * 8 + wave;
    if (tok >= NTOK) return;

    const float* px = x + (size_t)tok * D_DIM;
    float acc[E_DIM];
#pragma unroll
    for (int e = 0; e < E_DIM; ++e) acc[e] = 0.f;

    for (int d = lane; d < D_DIM; d += 32) {
        const float xv = px[d];
        const float* gw = gate_w + (size_t)d * E_DIM;
#pragma unroll
        for (int e = 0; e < E_DIM; ++e) acc[e] += xv * gw[e];
    }
#pragma unroll
    for (int off = 16; off >= 1; off >>= 1) {
#pragma unroll
        for (int e = 0; e < E_DIM; ++e)
            acc[e] += __shfl_xor(acc[e], off, 32);
    }

    if (lane == 0) {
        // softmax is monotonic: top-2 of probs == top-2 of logits
        float logit[E_DIM];
        int i1 = 0;
        logit[0] = acc[0] + gate_b[0];
        float v1 = logit[0];
#pragma unroll
        for (int e = 1; e < E_DIM; ++e) {
            logit[e] = acc[e] + gate_b[e];
            if (logit[e] > v1) { v1 = logit[e]; i1 = e; }   // ties: lowest idx
        }
        int i2 = -1; float v2 = -3.4e38f;
#pragma unroll
        for (int e = 0; e < E_DIM; ++e) {
            if (e == i1) continue;
            if (logit[e] > v2) { v2 = logit[e]; i2 = e; }
        }
        const int w = (i1 > i2) ? i1 : i2;   // largest expert index wins
        const int pos = atomicAdd(&counts[w], 1);
        perm[w * NTOK + pos] = tok;
    }
}

// ---------------------------------------------------------------------------
// Kernel 2: grouped expert GEMM on V_WMMA_F32_16X16X4_F32 (exact f32).
// Register-blocked: one wave owns a 64(token) x 32(O) tile = 4x2 WMMA tiles.
// Per K-step of 4: 4 A b64 + 4 B b32 loads feed 8 WMMAs (1.0 vmem/wmma).
// Scatter-accumulates (out+bias)/M into pooled[B,O] with f32 global atomics.
// ---------------------------------------------------------------------------
__global__ __launch_bounds__(32) void expert_kernel(
    const float* __restrict__ x, const float* __restrict__ expert_w,
    const float* __restrict__ expert_b, const int* __restrict__ counts,
    const int* __restrict__ perm, float* __restrict__ pooled)
{
    const int e   = blockIdx.z;
    const int cnt = counts[e];
    const int t0  = blockIdx.y * 64;
    if (t0 >= cnt) return;                 // early-exit null tiles
    const int o0   = blockIdx.x * 32;
    const int lane = threadIdx.x;          // wave32
    const int half = lane >> 4;            // selects K pair {0,1} vs {2,3}
    const int l16  = lane & 15;

    // A-matrix row pointers for the four 16-row groups (16x4 f32 layout:
    // lanes 0-15 = rows, lane half selects the K pair).
    const float* pa[4];
#pragma unroll
    for (int rg = 0; rg < 4; ++rg) {
        const int r  = t0 + rg * 16 + l16;
        const int tk = perm[e * NTOK + (r < cnt ? r : t0)];
        pa[rg] = x + (size_t)tk * D_DIM + half * 2;          // 8B aligned
    }

    // B-matrix (4x16 f32 layout: lanes = N cols, lane half selects K pair)
    // for the two 16-col groups.
    const float* pw  = expert_w + (size_t)e * D_DIM * O_DIM
                     + (size_t)(half * 2) * O_DIM;
    const float* pb0 = pw + o0 + l16;
    const float* pb1 = pw + o0 + 16 + l16;

    v8f c[4][2] = {};
    for (int kk = 0; kk < D_DIM; kk += 4) {
        v2f a[4];
#pragma unroll
        for (int rg = 0; rg < 4; ++rg)
            a[rg] = *(const v2f*)(pa[rg] + kk);
        v2f b0, b1;
        b0.x = pb0[(size_t)kk * O_DIM];
        b0.y = pb0[(size_t)kk * O_DIM + O_DIM];
        b1.x = pb1[(size_t)kk * O_DIM];
        b1.y = pb1[(size_t)kk * O_DIM + O_DIM];
#pragma unroll
        for (int rg = 0; rg < 4; ++rg) {
            c[rg][0] = __builtin_amdgcn_wmma_f32_16x16x4_f32(
                false, a[rg], false, b0, (short)0, c[rg][0], false, false);
            c[rg][1] = __builtin_amdgcn_wmma_f32_16x16x4_f32(
                false, a[rg], false, b1, (short)0, c[rg][1], false, false);
        }
    }

    // C/D layout: VGPR v holds row (v + 8*half), col = lane&15.
    const int   col0  = o0 + l16;
    const int   col1  = o0 + 16 + l16;
    const float bias0 = expert_b[e * O_DIM + col0];
    const float bias1 = expert_b[e * O_DIM + col1];
#pragma unroll
    for (int rg = 0; rg < 4; ++rg) {
#pragma unroll
        for (int v = 0; v < 8; ++v) {
            const int row = t0 + rg * 16 + half * 8 + v;
            if (row < cnt) {
                const int tokr = perm[e * NTOK + row];
                const int brow = tokr & (B_DIM - 1);          // tok % B
                float* prow = pooled + (size_t)brow * O_DIM;
                unsafeAtomicAdd(prow + col0, (c[rg][0][v] + bias0) * 0.25f);
                unsafeAtomicAdd(prow + col1, (c[rg][1][v] + bias1) * 0.25f);
            }
        }
    }
}

// ---------------------------------------------------------------------------
// Kernel 3: head GEMV [B,O] @ [O,2] + bias. One wave per batch row.
// ---------------------------------------------------------------------------
__global__ __launch_bounds__(256) void head_kernel(
    const float* __restrict__ pooled, const float* __restrict__ head_w,
    const float* __restrict__ head_b, float* __restrict__ out)
{
    const int lane = threadIdx.x & 31;
    const int wave = threadIdx.x >> 5;
    const int b = blockIdx.x * 8 + wave;
    if (b >= B_DIM) return;

    const float* pp = pooled + (size_t)b * O_DIM;
    float a0 = 0.f, a1 = 0.f;
    for (int o = lane; o < O_DIM; o += 32) {
        const float pv = pp[o];
        a0 += pv * head_w[o * C_DIM + 0];
        a1 += pv * head_w[o * C_DIM + 1];
    }
#pragma unroll
    for (int off = 16; off >= 1; off >>= 1) {
        a0 += __shfl_xor(a0, off, 32);
        a1 += __shfl_xor(a1, off, 32);
    }
    if (lane == 0) {
        out[b * C_DIM + 0] = a0 + head_b[0];
        out[b * C_DIM + 1] = a1 + head_b[1];
    }
}

// ---------------------------------------------------------------------------
extern "C" void kernel_launch(void* const* d_in, const int* in_sizes, int n_in,
                              void* d_out, int out_size, void* d_ws, size_t ws_size,
                              hipStream_t stream)
{
    const float* x        = (const float*)d_in[0];
    const float* gate_w   = (const float*)d_in[1];
    const float* gate_b   = (const float*)d_in[2];
    const float* expert_w = (const float*)d_in[3];
    const float* expert_b = (const float*)d_in[4];
    const float* head_w   = (const float*)d_in[5];
    const float* head_b   = (const float*)d_in[6];
    float* out = (float*)d_out;

    // Workspace layout: pooled[B,O] f32 (8 MB) | counts[E] | perm[E][NTOK]
    const size_t pooled_bytes = (size_t)B_DIM * O_DIM * sizeof(float);
    float* pooled = (float*)d_ws;
    int*   counts = (int*)((char*)d_ws + pooled_bytes);
    int*   perm   = counts + E_DIM;

    // Zero pooled accumulator + expert counts (capture-safe memset node).
    hipMemsetAsync(d_ws, 0, pooled_bytes + E_DIM * sizeof(int), stream);

    gate_kernel<<<NTOK / 8, 256, 0, stream>>>(x, gate_w, gate_b, counts, perm);

    dim3 grid(O_DIM / 32, NTOK / 64, E_DIM);   // 32 x 128 x 8, 1 wave each
    expert_kernel<<<grid, 32, 0, stream>>>(x, expert_w, expert_b,
                                           counts, perm, pooled);

    head_kernel<<<B_DIM / 8, 256, 0, stream>>>(pooled, head_w, head_b, out);
}